// LIFCell_69106023792954
// MI455X (gfx1250) — compile-verified
//
#include <hip/hip_runtime.h>
#include <stdint.h>

// LIF scan: v = alpha*v + I[t]; s = (v>=1); v = s ? 0 : v
// I: [B=64, T=512, N=1024] f32.  Outputs: spikes [B,T,N], vtraj [B,T,N] (concat in d_out).
//
// Memory-bound streaming recurrence: 402 MB traffic -> ~17us floor at 23.3 TB/s.
// Parallelism capped at B*N = 65536 chains (2048 wave32). CDNA5 async
// global->LDS pipeline (ASYNCcnt) keeps a depth-32 per-thread input ring in
// LDS: 32 outstanding 128B loads per wave (~8 MB in flight device-wide) with
// zero VGPR cost and no barriers (each thread reads only its own ring column;
// async loads complete in order, so s_wait_asynccnt gates the oldest slot).

#define B_LEN   64
#define T_LEN   512
#define N_LEN   1024
#define BLOCK   256
#define DEPTH   32            // ring depth (power of 2, <= 63 for ASYNCcnt)

// Async 4-byte global->LDS copy, GVS form: 64-bit uniform SGPR base +
// 32-bit per-lane voffset (avoids 64-bit per-lane address arithmetic).
// VDST = LDS byte offset, VADDR = 32-bit byte offset, SADDR = base.
__device__ __forceinline__ void async_ld_b32(uint32_t voff_bytes,
                                             const float* base_uniform,
                                             uint32_t lds_off) {
    asm volatile("global_load_async_to_lds_b32 %0, %1, %2"
                 :: "v"(lds_off), "v"(voff_bytes), "s"(base_uniform)
                 : "memory");
}

__global__ __launch_bounds__(BLOCK) void lif_scan_kernel(
    const float* __restrict__ I,
    float*       __restrict__ spk,
    float*       __restrict__ vtr)
{
    __shared__ float ring[DEPTH][BLOCK];

    const int tid = threadIdx.x;
    const int b   = blockIdx.y;                       // 0..63

    // Block-uniform global base for this (b, n-block); per-lane part is 32-bit.
    const size_t ubase = (size_t)b * T_LEN * N_LEN + (size_t)blockIdx.x * BLOCK;
    const float* gin_base = I + ubase;                // uniform -> SGPR pair
    const uint32_t voff0  = (uint32_t)tid * 4u;       // per-lane byte offset
    const uint32_t tstride = (uint32_t)N_LEN * 4u;    // 4096 B per time step

    // Per-lane output pointers (compiler picks addressing; stores are NT).
    float* gs = spk + ubase + tid;
    float* gv = vtr + ubase + tid;

    // Flat shared pointer -> LDS byte offset (LDS offset is the low 32 bits of the address).
    const uint32_t lds_base = (uint32_t)(uintptr_t)&ring[0][tid];

    // Prime the ring: DEPTH outstanding async loads.
#pragma unroll
    for (int k = 0; k < DEPTH; ++k) {
        async_ld_b32(voff0 + (uint32_t)k * tstride, gin_base,
                     lds_base + (uint32_t)k * (BLOCK * 4));
    }

    const float alpha = 0.6065306597126334f;          // exp(-1/2)
    float v = 0.0f;

    // Steady state: wait oldest slot, consume, refill for t+DEPTH.
#pragma unroll 4
    for (int t = 0; t < T_LEN - DEPTH; ++t) {
        // completed >= t+1  <=>  ASYNCcnt <= DEPTH-1 (in-order completion)
        asm volatile("s_wait_asynccnt %0" :: "i"(DEPTH - 1) : "memory");

        const int slot = t & (DEPTH - 1);
        float i_t = ring[slot][tid];                  // ds_load_b32, own column

        v = fmaf(alpha, v, i_t);
        float s = (v >= 1.0f) ? 1.0f : 0.0f;
        v       = (v >= 1.0f) ? 0.0f : v;             // reset to V_RESET = 0

        // Refill this slot for step t+DEPTH (after the ds_load consumed it).
        async_ld_b32(voff0 + (uint32_t)(t + DEPTH) * tstride, gin_base,
                     lds_base + (uint32_t)slot * (BLOCK * 4));

        // Write-once 256MB streams: nontemporal, don't thrash the 192MB L2.
        __builtin_nontemporal_store(s, gs + (size_t)t * N_LEN);
        __builtin_nontemporal_store(v, gv + (size_t)t * N_LEN);
    }

    // Drain: all T loads issued; wait for the rest, finish last DEPTH steps.
    asm volatile("s_wait_asynccnt 0" ::: "memory");

#pragma unroll 4
    for (int t = T_LEN - DEPTH; t < T_LEN; ++t) {
        float i_t = ring[t & (DEPTH - 1)][tid];

        v = fmaf(alpha, v, i_t);
        float s = (v >= 1.0f) ? 1.0f : 0.0f;
        v       = (v >= 1.0f) ? 0.0f : v;

        __builtin_nontemporal_store(s, gs + (size_t)t * N_LEN);
        __builtin_nontemporal_store(v, gv + (size_t)t * N_LEN);
    }
}

extern "C" void kernel_launch(void* const* d_in, const int* in_sizes, int n_in,
                              void* d_out, int out_size, void* d_ws, size_t ws_size,
                              hipStream_t stream) {
    (void)in_sizes; (void)n_in; (void)d_ws; (void)ws_size; (void)out_size;

    const float* I   = (const float*)d_in[0];
    float*       out = (float*)d_out;
    const size_t total = (size_t)B_LEN * T_LEN * N_LEN;

    float* spikes = out;            // first output
    float* vtraj  = out + total;    // second output

    dim3 grid(N_LEN / BLOCK, B_LEN);   // (4, 64) blocks of 256 = 65536 chains
    lif_scan_kernel<<<grid, BLOCK, 0, stream>>>(I, spikes, vtraj);
}